// JAXEPG_15753940042261
// MI455X (gfx1250) — compile-verified
//
#include <hip/hip_runtime.h>

#define NS 50
#define NP 200
#define BLK 128

__global__ __launch_bounds__(BLK, 1)
void epg_spgr_kernel(const float* __restrict__ T1,
                     const float* __restrict__ T2,
                     const float* __restrict__ alpha,
                     const float* __restrict__ TRp,
                     float* __restrict__ out, int n)
{
    const int tid = threadIdx.x;
    const int i   = blockIdx.x * BLK + tid;
    const int ci  = (i < n) ? i : (n - 1);   // clamped index for safe staging

    // gfx1250 prefetch path (global_prefetch_b8)
    __builtin_prefetch(T1 + ci, 0, 0);
    __builtin_prefetch(T2 + ci, 0, 0);
    __builtin_prefetch(alpha + ci, 0, 0);

    float t1, t2, a;
#if defined(__HIP_DEVICE_COMPILE__) && defined(__gfx1250__)
    // CDNA5 async copy: global -> LDS, tracked by ASYNCcnt (wave-local wait,
    // each lane touches only its own slot -> no barrier needed).
    __shared__ float stage[3 * BLK];
    {
        unsigned l0 = (unsigned)(unsigned long long)(&stage[tid]);
        unsigned l1 = (unsigned)(unsigned long long)(&stage[BLK + tid]);
        unsigned l2 = (unsigned)(unsigned long long)(&stage[2 * BLK + tid]);
        const float* g0 = T1 + ci;
        const float* g1 = T2 + ci;
        const float* g2 = alpha + ci;
        asm volatile(
            "global_load_async_to_lds_b32 %0, %3, off\n\t"
            "global_load_async_to_lds_b32 %1, %4, off\n\t"
            "global_load_async_to_lds_b32 %2, %5, off\n\t"
            "s_wait_asynccnt 0x0"
            :
            : "v"(l0), "v"(l1), "v"(l2), "v"(g0), "v"(g1), "v"(g2)
            : "memory");
    }
    t1 = stage[tid];
    t2 = stage[BLK + tid];
    a  = stage[2 * BLK + tid];
#else
    t1 = T1[ci];
    t2 = T2[ci];
    a  = alpha[ci];
#endif
    if (i >= n) return;

    const float TR = TRp[0];
    const float E1 = expf(-TR / t1);
    const float E2 = expf(-TR / t2);
    const float ch = cosf(0.5f * a);
    const float sh = sinf(0.5f * a);
    const float s2 = sh * sh;
    const float sa = sinf(a);
    const float ca = cosf(a);
    const float oneME1 = 1.0f - E1;

    // Loop-invariant relaxation-folded coefficient factors
    const float c2E2  = (ch * ch) * E2;     // c2 * E2
    const float s2E2  = s2 * E2;            // s2 * E2
    const float saE2  = sa * E2;            // sa * E2
    const float caE1  = ca * E1;            // ca * E1
    const float saE1h = 0.5f * sa * E1;     // 0.5 * sa * E1

    // Register-resident EPG state: Fp, Fm, Z (complex), 50 orders each.
    float Fpr[NS], Fpi[NS], Fmr[NS], Fmi[NS], Zr[NS], Zi[NS];
#pragma unroll
    for (int k = 0; k < NS; ++k) {
        Fpr[k] = 0.0f; Fpi[k] = 0.0f;
        Fmr[k] = 0.0f; Fmi[k] = 0.0f;
        Zr[k]  = 0.0f; Zi[k]  = 0.0f;
    }
    Zr[0] = 1.0f;

    const float SPOIL = 2.0420352248333655f; // float32(deg2rad(117))
    float phi = 0.0f, inc = 0.0f;
    float sig_re = 0.0f, sig_im = 0.0f;      // E2-scaled Fp_rf[0]

    for (int p = 0; p < NP; ++p) {
        float sp, cp;
        sincosf(phi, &sp, &cp);               // ep = cp + i*sp
        const float c2p = cp * cp - sp * sp;  // cos(2*phi)
        const float s2p = 2.0f * cp * sp;     // sin(2*phi)
        // Relaxation-folded per-pulse coefficients:
        const float ArE = c2p * s2E2;         // Re(ep^2*s2)*E2
        const float AiE = s2p * s2E2;         // Im(ep^2*s2)*E2
        const float sCE = sp * saE2;          // |.. sa|*E2 pieces for C/D rows
        const float cCE = cp * saE2;
        const float sZE = sp * saE1h;         // 0.5*sa*E1 pieces for Z row
        const float cZE = cp * saE1h;

        float carry_re = 0.0f, carry_im = 0.0f;  // Fp_r[k-1] carry (shift up)
        float fm1_re = 0.0f, fm1_im = 0.0f;      // Fm_r[1] (feeds Fp_s[0], Fm_s[0])

#pragma unroll
        for (int k = 0; k < NS; ++k) {
            const float pr  = Fpr[k], pi_ = Fpi[k];
            const float mr  = Fmr[k], mi  = Fmi[k];
            const float zre = Zr[k],  zim = Zi[k];

            // RF rotation with relaxation pre-folded (5 FMA ops per component)
            const float fp_re = c2E2 * pr  + ArE * mr - AiE * mi  + sCE * zre + cCE * zim;
            const float fp_im = c2E2 * pi_ + ArE * mi + AiE * mr  + sCE * zim - cCE * zre;
            const float fm_re = ArE * pr + AiE * pi_ + c2E2 * mr  + sCE * zre - cCE * zim;
            const float fm_im = ArE * pi_ - AiE * pr + c2E2 * mi  + sCE * zim + cCE * zre;
            const float zn_re = -sZE * pr + cZE * pi_ - sZE * mr - cZE * mi + caE1 * zre;
            const float zn_im = -sZE * pi_ - cZE * pr - sZE * mi + cZE * mr + caE1 * zim;

            if (k == 0) { sig_re = fp_re; sig_im = fp_im; } // == E2*Fp_rf[0]

            Zr[k] = (k == 0) ? (zn_re + oneME1) : zn_re;
            Zi[k] = zn_im;

            // Gradient shift (in-place):
            // Fm_s[k] = Fm_r[k+1] -> write Fm[k-1] while ascending
            if (k >= 1) { Fmr[k - 1] = fm_re; Fmi[k - 1] = fm_im; }
            if (k == 1) { fm1_re = fm_re; fm1_im = fm_im; }
            // Fp_s[k] = Fp_r[k-1] -> one-element carry
            if (k >= 1) { Fpr[k] = carry_re; Fpi[k] = carry_im; }
            carry_re = fp_re; carry_im = fp_im;
        }
        // Reference: Fp_s[0] = Fm_r[1] (no conjugate -- replicated exactly); Fm_s[49] = 0
        Fpr[0] = fm1_re; Fpi[0] = fm1_im;
        Fmr[NS - 1] = 0.0f; Fmi[NS - 1] = 0.0f;

        inc += SPOIL;   // same fp32 add order as the reference scan
        phi += inc;
    }

    // |sig| = |E2*Fp_rf[0]| / E2  (|exp(-i*phi)| == 1)
    out[i] = sqrtf(sig_re * sig_re + sig_im * sig_im) / E2;
}

extern "C" void kernel_launch(void* const* d_in, const int* in_sizes, int n_in,
                              void* d_out, int out_size, void* d_ws, size_t ws_size,
                              hipStream_t stream) {
    const float* T1 = (const float*)d_in[0];
    const float* T2 = (const float*)d_in[1];
    const float* al = (const float*)d_in[2];
    const float* TR = (const float*)d_in[3];
    float* out = (float*)d_out;
    const int n = in_sizes[0];

    const int grid = (n + BLK - 1) / BLK;
    hipLaunchKernelGGL(epg_spgr_kernel, dim3(grid), dim3(BLK), 0, stream,
                       T1, T2, al, TR, out, n);
}